// SelfAttention_36790689858288
// MI455X (gfx1250) — compile-verified
//
#include <hip/hip_runtime.h>

typedef float v2f __attribute__((ext_vector_type(2)));
typedef float v8f __attribute__((ext_vector_type(8)));
typedef unsigned int u32x4 __attribute__((ext_vector_type(4)));
typedef int i32x8 __attribute__((ext_vector_type(8)));
typedef int i32x4 __attribute__((ext_vector_type(4)));

#if defined(__gfx1250__) && __has_builtin(__builtin_amdgcn_tensor_load_to_lds) && \
    __has_builtin(__builtin_amdgcn_s_wait_tensorcnt)
#define USE_TDM 1
#else
#define USE_TDM 0
#endif

// ---------------------------------------------------------------------------
// WMMA f32 GEMM:  C[m,n] = sum_k A[m,k] * Wt[n,k]  (+ bias[n])
//   A : (M, K) row-major, Wt : (Nd, K) row-major (i.e. B = Wt^T)
// One wave per 16x16 C tile, 8 waves (8 M-tiles) per block sharing one
// 16-row Wt panel staged into LDS by the Tensor Data Mover (TDM).
// LDS panel layout: row-major with +1 DWORD pad per row (TDM pad_enable),
// stride K+1 floats -> (K+1) mod 64 == 1 (K=128/256) => conflict-free B
// fetches, and {k,k+1} pairs are contiguous => single ds_load_b64.
// V_WMMA_F32_16X16X4_F32 operand layout (ISA 7.12.2):
//   A 16x4 : lane<16 -> {K=0,K=1}, lane>=16 -> {K=2,K=3}, M = lane&15
//   B 4x16 : mirrored halves, N = lane&15
//   C/D    : 8 VGPRs, row = r + (lane<16 ? 0 : 8), col = lane&15
// ---------------------------------------------------------------------------
template <int K>
__global__ __launch_bounds__(256) void wmma_gemm_wt(
    const float* __restrict__ A, const float* __restrict__ Wt,
    const float* __restrict__ bias, float* __restrict__ C, int M, int Nd) {
  constexpr int S = K + 1;          // padded LDS row stride (floats)
  __shared__ float ldsB[16 * S];    // ldsB[n*S + k] = Wt[n0+n][k]
  const int tid = threadIdx.x;
  const int lane = tid & 31;
  const int wave = tid >> 5;
  const int n0 = blockIdx.y * 16;

#if USE_TDM
  if (wave == 0) {
    // ---- Tensor DMA descriptor (D#), CDNA5 ISA ----
    // pad_interval code: interval = 2^(code+1) DWORDs; we want K.
    constexpr unsigned padc = (K == 32) ? 4u : (K == 128) ? 6u : 7u;
    const unsigned long long ga =
        (unsigned long long)(size_t)(Wt + (size_t)n0 * K);
    u32x4 g0;
    g0[0] = 1u;                                   // count=1, is_restore=0
    g0[1] = (unsigned)(size_t)ldsB;               // lds_addr (bytes)
    g0[2] = (unsigned)ga;                         // global_addr[31:0]
    g0[3] = (unsigned)((ga >> 32) & 0x1FFFFFFu)   // global_addr[56:32]
            | (2u << 30);                         // type = 2 ("image")
    i32x8 g1;
    g1[0] = (int)((2u << 16)                      // data_size = 4 bytes
                  | (1u << 20)                    // pad_enable
                  | (padc << 22));                // pad_interval = K DWORDs
                                                  // pad_amount code 0 = 1 DWORD
    g1[1] = (int)(((unsigned)K & 0xFFFFu) << 16); // tensor_dim0[15:0] = K
    g1[2] = (int)(16u << 16);                     // tensor_dim1 = 16 rows
    g1[3] = (int)((unsigned)K << 16);             // tile_dim0 = K
    g1[4] = 16;                                   // tile_dim1 = 16, tile_dim2 = 0
    g1[5] = K;                                    // tensor_dim0_stride = K
    g1[6] = 0;
    g1[7] = 0;
    const i32x4 gz4 = {0, 0, 0, 0};               // 2D tensor: groups 2/3 unused
    const i32x8 gz8 = {0, 0, 0, 0, 0, 0, 0, 0};   // clang-23 6-arg form
    __builtin_amdgcn_tensor_load_to_lds(g0, g1, gz4, gz4, gz8, 0);
    __builtin_amdgcn_s_wait_tensorcnt(0);         // TDM done -> LDS valid
  }
#else
  // Fallback staging (same padded row-major layout), coalesced global reads.
#pragma unroll 4
  for (int i = tid; i < 16 * K; i += 256) {
    const int n = i / K;
    const int k = i % K;
    ldsB[n * S + k] = Wt[(n0 + n) * K + k];
  }
#endif
  __syncthreads();

  const int mTile = blockIdx.x * 8 + wave;
  if (mTile * 16 >= M) return;  // tail guard (after barrier)

  const int m0 = mTile * 16;
  const int mr = lane & 15;                 // M row (A) / N col (B,D)
  const int koff = (lane < 16) ? 0 : 2;     // K sub-offset per lane half
  const float* Arow = A + (long)(m0 + mr) * K + koff;
  const float* Brow = ldsB + mr * S + koff;

  v8f acc = {};
#pragma unroll
  for (int kb = 0; kb < K; kb += 4) {
    v2f a, b;
    a.x = Arow[kb];
    a.y = Arow[kb + 1];
    b.x = Brow[kb];           // contiguous pair -> ds_load_b64
    b.y = Brow[kb + 1];
    acc = __builtin_amdgcn_wmma_f32_16x16x4_f32(
        /*neg_a=*/false, a, /*neg_b=*/false, b,
        /*c_mod=*/(short)0, acc, /*reuse_a=*/false, /*reuse_b=*/false);
  }

  const int rbase = (lane < 16) ? 0 : 8;
  const int col = n0 + mr;
  const float bv = bias ? bias[col] : 0.0f;
#pragma unroll
  for (int r = 0; r < 8; ++r)
    C[(long)(m0 + rbase + r) * Nd + col] = acc[r] + bv;
}

// ---------------------------------------------------------------------------
// Windowed attention: one block per (n,l); 8 waves = 8 heads.
// Gathered key/value rows stay in L2 (k_proj per batch ~10MB << 192MB L2).
// Wave32 xor-shuffle reductions for dot products / softmax max / softmax sum.
// idx == 600 is the pad sentinel (energy = -1e20, value row = 0).
// ---------------------------------------------------------------------------
__global__ __launch_bounds__(256) void attn_window_kernel(
    const float* __restrict__ qp, const float* __restrict__ kp,
    const float* __restrict__ vp, const int* __restrict__ mask,
    float* __restrict__ oa, float* __restrict__ attn_last) {
  const int nl = blockIdx.x;        // n*600 + l
  const int n = nl / 600;
  const int tid = threadIdx.x;
  const int lane = tid & 31;
  const int h = tid >> 5;           // head == wave id

  __shared__ int s_idx[64];
  __shared__ float s_p[8][64];

  if (tid < 64) s_idx[tid] = mask[nl * 64 + tid];
  __syncthreads();

  // q row for this head: 128 floats -> float4 per lane (coalesced)
  const float4* q4 = (const float4*)(qp + ((long)nl * 8 + h) * 128);
  const float4 qv = q4[lane];

  // ---- energies over the 64-wide window ----
  float e0 = -1e20f, e1 = -1e20f;
  for (int w = 0; w < 64; ++w) {
    const int idx = s_idx[w];
    const int row = idx < 600 ? idx : 599;
    const float4* k4 = (const float4*)(kp + (((long)n * 600 + row) * 8 + h) * 128);
    const float4 kv = k4[lane];
    float part = qv.x * kv.x + qv.y * kv.y + qv.z * kv.z + qv.w * kv.w;
#pragma unroll
    for (int off = 16; off >= 1; off >>= 1)
      part += __shfl_xor(part, off, 32);
    const float ev = (idx >= 600) ? -1e20f : part;
    if (w == lane) e0 = ev;
    if (w == lane + 32) e1 = ev;
  }

  // ---- softmax across the 64 window slots (2 per lane) ----
  float mx = fmaxf(e0, e1);
#pragma unroll
  for (int off = 16; off >= 1; off >>= 1)
    mx = fmaxf(mx, __shfl_xor(mx, off, 32));
  float p0 = __expf(e0 - mx);
  float p1 = __expf(e1 - mx);
  float s = p0 + p1;
#pragma unroll
  for (int off = 16; off >= 1; off >>= 1)
    s += __shfl_xor(s, off, 32);
  const float inv = 1.0f / s;
  p0 *= inv;
  p1 *= inv;
  s_p[h][lane] = p0;
  s_p[h][lane + 32] = p1;
  if (h == 7) {  // reference returns attn[:, :, -1, :]
    attn_last[(long)nl * 64 + lane] = p0;
    attn_last[(long)nl * 64 + lane + 32] = p1;
  }

  // ---- weighted value sum: Dv = 32 -> lane == output channel ----
  float acc = 0.0f;
  for (int w = 0; w < 64; ++w) {
    const int idx = s_idx[w];
    const int row = idx < 600 ? idx : 599;
    float vv = vp[(((long)n * 600 + row) * 8 + h) * 32 + lane];
    vv = (idx >= 600) ? 0.0f : vv;     // pad rows are zero in the reference
    acc += s_p[h][w] * vv;             // same-wave LDS: DS ops are in-order
  }
  oa[(long)nl * 256 + h * 32 + lane] = acc;
}

// ---------------------------------------------------------------------------
// Launch: 3 WMMA projections -> attention -> WMMA output projection (+bias)
// Workspace (floats): qp 19200*128 | kp 19200*128 | vp 19200*32 | oa 2400*256
//   total = 24.6 MB
// ---------------------------------------------------------------------------
extern "C" void kernel_launch(void* const* d_in, const int* in_sizes, int n_in,
                              void* d_out, int out_size, void* d_ws,
                              size_t ws_size, hipStream_t stream) {
  const float* value = (const float*)d_in[0];
  const float* key   = (const float*)d_in[1];
  const float* query = (const float*)d_in[2];
  const int*   mask  = (const int*)d_in[3];
  const float* Wv = (const float*)d_in[4];
  const float* Wk = (const float*)d_in[5];
  const float* Wq = (const float*)d_in[6];
  const float* Wo = (const float*)d_in[7];
  const float* bo = (const float*)d_in[8];

  float* out = (float*)d_out;                 // (4,600,256)
  float* attn_last = out + 4 * 600 * 256;     // (4,600,64)

  float* qp = (float*)d_ws;                   // (19200,128)
  float* kp = qp + 19200 * 128;               // (19200,128)
  float* vp = kp + 19200 * 128;               // (19200,32)
  float* oa = vp + 19200 * 32;                // (2400,256)

  const dim3 blk(256);
  // Q/K projections: M=19200 -> 1200 tiles -> 150 blocks; Nd=128 -> grid.y=8
  wmma_gemm_wt<128><<<dim3(150, 8), blk, 0, stream>>>(query, Wq, nullptr, qp, 19200, 128);
  wmma_gemm_wt<128><<<dim3(150, 8), blk, 0, stream>>>(key,   Wk, nullptr, kp, 19200, 128);
  // V projection: Nd=32 -> grid.y=2
  wmma_gemm_wt<32><<<dim3(150, 2), blk, 0, stream>>>(value, Wv, nullptr, vp, 19200, 32);
  // Attention: one block per (n,l)
  attn_window_kernel<<<dim3(2400), blk, 0, stream>>>(qp, kp, vp, mask, oa, attn_last);
  // Output projection: M=2400 -> 150 tiles -> 19 blocks (tail guarded); Nd=256
  wmma_gemm_wt<256><<<dim3(19, 16), blk, 0, stream>>>(oa, Wo, bo, out, 2400, 256);
}